// CandidateAwareModel_79139067396557
// MI455X (gfx1250) — compile-verified
//
#include <hip/hip_runtime.h>
#include <math.h>

#define B_ 2
#define S_ 128
#define H_ 768
#define C_ 5
constexpr float NEGV = -1024.0f;
constexpr int M_   = B_ * S_;        // 256 rows
constexpr int N_   = H_ + C_ * H_;   // 4608 cols (h_p | h_a)
constexpr int NT_N = N_ / 16;        // 288
constexpr int NT_M = M_ / 16;        // 16

typedef float v2f __attribute__((ext_vector_type(2)));
typedef float v8f __attribute__((ext_vector_type(8)));

// Native gfx1250 V_TANH_F32 (TRANS pipe). Prefer the clang builtin when it
// exists (best scheduling); otherwise force the instruction via inline asm.
#if __has_builtin(__builtin_amdgcn_tanhf)
__device__ __forceinline__ float tanh_hw(float x) { return __builtin_amdgcn_tanhf(x); }
#elif __has_builtin(__builtin_amdgcn_tanh_f32)
__device__ __forceinline__ float tanh_hw(float x) { return __builtin_amdgcn_tanh_f32(x); }
#else
__device__ __forceinline__ float tanh_hw(float x) {
    float r;
    asm volatile("v_tanh_f32 %0, %1" : "=v"(r) : "v"(x));
    return r;
}
#endif
#define TANHF(x) tanh_hw(x)

// ---------------------------------------------------------------------------
// K0: mask addend (0 or NEG) per (b, p, a); also zero the loss accumulators.
// mask[b,p,a] = (attn[b,a] > 0) && OR_c ng[b,p,c,a]
// ---------------------------------------------------------------------------
__global__ void mask_init_kernel(const int* __restrict__ attn,
                                 const int* __restrict__ ng,
                                 float* __restrict__ madd,
                                 float* __restrict__ accum) {
    int idx = blockIdx.x * blockDim.x + threadIdx.x;
    if (idx == 0) { accum[0] = 0.0f; accum[1] = 0.0f; }
    if (idx >= B_ * S_ * S_) return;
    int a = idx % S_;
    int p = (idx / S_) % S_;
    int b = idx / (S_ * S_);
    bool any = false;
    if (attn[b * S_ + a] > 0) {
#pragma unroll
        for (int c = 0; c < C_; ++c)
            any = any || (ng[((b * S_ + p) * C_ + c) * S_ + a] != 0);
    }
    madd[idx] = any ? 0.0f : NEGV;
}

// ---------------------------------------------------------------------------
// K1: fp32 GEMM via V_WMMA_F32_16X16X4_F32.
// h_cat(256 x 4608) = X(256 x 768) @ [W_prd; W_arg]^T + [b_prd; b_arg]
// One wave per 16x16 tile, K=768 in steps of 4.
// A frag: lanes 0-15 row M=lane, K={k,k+1}; lanes 16-31 same rows, K={k+2,k+3}
// B frag: same with M->N (B[k][n] = W[n][k], contiguous along k in W row-major)
// ---------------------------------------------------------------------------
__global__ void __launch_bounds__(128)
gemm_wmma_kernel(const float* __restrict__ seq,
                 const float* __restrict__ Wprd, const float* __restrict__ bprd,
                 const float* __restrict__ Warg, const float* __restrict__ barg,
                 float* __restrict__ hcat) {
    const int lane = threadIdx.x & 31;
    const int wave = threadIdx.x >> 5;
    const int tile = blockIdx.x * 4 + wave;        // 0 .. 4607
    const int tM = tile / NT_N;
    const int tN = tile % NT_N;
    const int l16  = lane & 15;
    const int grp2 = (lane >> 4) << 1;             // 0 or 2

    const int n = tN * 16 + l16;
    const float* bp;
    float bias;
    if (n < H_) { bp = Wprd + (size_t)n * H_;        bias = bprd[n]; }
    else        { bp = Warg + (size_t)(n - H_) * H_; bias = barg[n - H_]; }
    bp += grp2;
    const float* ap = seq + (size_t)(tM * 16 + l16) * H_ + grp2;

    v8f acc = {};
    for (int k = 0; k < H_; k += 4) {
        v2f av = *(const v2f*)(ap + k);
        v2f bv = *(const v2f*)(bp + k);
        acc = __builtin_amdgcn_wmma_f32_16x16x4_f32(
            /*neg_a=*/false, av, /*neg_b=*/false, bv,
            /*c_mod=*/(short)0, acc, /*reuse_a=*/false, /*reuse_b=*/false);
    }

    // D layout: VGPR r -> M = r (+8 for lanes 16-31), N = lane&15
    const int mbase = tM * 16 + ((lane >> 4) << 3);
#pragma unroll
    for (int r = 0; r < 8; ++r)
        hcat[(size_t)(mbase + r) * N_ + n] = acc[r] + bias;
}

// ---------------------------------------------------------------------------
// K2: biaffine tanh contraction.
// out[b,p,c,a] = sum_h tanh(hp[b,p,h] + ha[b,a,c,h]) * Wout[c,h]  + mask addend
// Block = 256 threads = one 16(p) x 16(a) tile for fixed (b,c).
// H staged through LDS in chunks of 256 (stride padded to 260 -> no bank
// conflicts on the 16-row ha reads; float4 ds_load_b128 inner loop).
// Inner loop per 4 h: 3x ds_load_b128 + 4x (v_add_f32, v_tanh_f32, v_fma).
// ---------------------------------------------------------------------------
#define CH  256
#define CHP 260
__global__ void __launch_bounds__(256)
biaffine_kernel(const float* __restrict__ hcat, const float* __restrict__ Wout,
                const float* __restrict__ madd, float* __restrict__ out) {
    __shared__ float hp[16][CHP];
    __shared__ float ha[16][CHP];
    __shared__ float wc[CH];

    const int bid = blockIdx.x;
    const int at = bid & 7;
    const int pt = (bid >> 3) & 7;
    const int c  = (bid >> 6) % C_;
    const int b  = bid / (64 * C_);

    const int t   = threadIdx.x;
    const int p   = t >> 4;
    const int a   = t & 15;
    const int lr  = t >> 4;         // row this thread loads
    const int lcb = (t & 15) << 4;  // 16-float column slab

    float acc = 0.0f;
    for (int ch = 0; ch < H_ / CH; ++ch) {
        const float* srcp = hcat + (size_t)(b * S_ + pt * 16 + lr) * N_ + ch * CH + lcb;
        const float* srca = hcat + (size_t)(b * S_ + at * 16 + lr) * N_ + H_ + c * H_ + ch * CH + lcb;
#pragma unroll
        for (int i = 0; i < 4; ++i) {
            *(float4*)&hp[lr][lcb + 4 * i] = *(const float4*)(srcp + 4 * i);
            *(float4*)&ha[lr][lcb + 4 * i] = *(const float4*)(srca + 4 * i);
        }
        wc[t] = Wout[c * H_ + ch * CH + t];
        __syncthreads();

        for (int h = 0; h < CH; h += 4) {
            float4 xp = *(const float4*)&hp[p][h];
            float4 xa = *(const float4*)&ha[a][h];
            float4 w4 = *(const float4*)&wc[h];
            acc += TANHF(xp.x + xa.x) * w4.x;
            acc += TANHF(xp.y + xa.y) * w4.y;
            acc += TANHF(xp.z + xa.z) * w4.z;
            acc += TANHF(xp.w + xa.w) * w4.w;
        }
        __syncthreads();
    }
    const int pg = pt * 16 + p;
    const int ag = at * 16 + a;
    out[((size_t)(b * S_ + pg) * C_ + c) * S_ + ag] =
        acc + madd[(b * S_ + pg) * S_ + ag];
}

// ---------------------------------------------------------------------------
// K3: masked multi-hot cross-entropy. One wave32 per (b,p,c) row of 128.
// ---------------------------------------------------------------------------
__global__ void __launch_bounds__(256)
loss_kernel(const float* __restrict__ out, const float* __restrict__ target,
            float* __restrict__ accum) {
    const int gid  = blockIdx.x * blockDim.x + threadIdx.x;
    const int row  = gid >> 5;
    const int lane = gid & 31;
    if (row >= B_ * S_ * C_) return;

    const float* x  = out    + (size_t)row * S_;
    const float* tg = target + (size_t)row * S_;

    float v[4], tv[4];
    float m = -INFINITY;
#pragma unroll
    for (int i = 0; i < 4; ++i) {
        v[i]  = x[lane + 32 * i];
        tv[i] = tg[lane + 32 * i];
        m = fmaxf(m, v[i]);
    }
    for (int o = 16; o > 0; o >>= 1) m = fmaxf(m, __shfl_xor(m, o, 32));

    float s = 0.0f;
#pragma unroll
    for (int i = 0; i < 4; ++i) s += expf(v[i] - m);
    for (int o = 16; o > 0; o >>= 1) s += __shfl_xor(s, o, 32);

    const float lse = m + logf(s);
    float num = 0.0f, den = 0.0f;
#pragma unroll
    for (int i = 0; i < 4; ++i) {
        num += tv[i] * (lse - v[i]);   // -log_softmax * target
        den += tv[i];
    }
    for (int o = 16; o > 0; o >>= 1) {
        num += __shfl_xor(num, o, 32);
        den += __shfl_xor(den, o, 32);
    }
    if (lane == 0) {
        atomicAdd(&accum[0], num);
        atomicAdd(&accum[1], den);
    }
}

__global__ void finalize_kernel(const float* __restrict__ accum, float* __restrict__ loss) {
    loss[0] = accum[0] / accum[1];
}

// ---------------------------------------------------------------------------
extern "C" void kernel_launch(void* const* d_in, const int* in_sizes, int n_in,
                              void* d_out, int out_size, void* d_ws, size_t ws_size,
                              hipStream_t stream) {
    const float* seq    = (const float*)d_in[0];   // (B,S,H)
    const int*   attn   = (const int*)  d_in[1];   // (B,S)
    const int*   ng     = (const int*)  d_in[2];   // (B,S,C,S) bool->int
    const float* target = (const float*)d_in[3];   // (B,S,C,S)
    const float* Wprd   = (const float*)d_in[4];   // (H,H)
    const float* bprd   = (const float*)d_in[5];   // (H,)
    const float* Warg   = (const float*)d_in[6];   // (C*H,H)
    const float* barg   = (const float*)d_in[7];   // (C*H,)
    const float* Wout   = (const float*)d_in[8];   // (C,H)
    float* out = (float*)d_out;                    // [loss | (B,S,C,S)]

    float* hcat  = (float*)d_ws;                   // 256*4608 floats
    float* madd  = hcat + (size_t)M_ * N_;         // B*S*S floats
    float* accum = madd + B_ * S_ * S_;            // 2 floats

    mask_init_kernel<<<(B_ * S_ * S_ + 255) / 256, 256, 0, stream>>>(attn, ng, madd, accum);
    gemm_wmma_kernel<<<(NT_M * NT_N) / 4, 128, 0, stream>>>(seq, Wprd, bprd, Warg, barg, hcat);
    biaffine_kernel<<<B_ * C_ * 8 * 8, 256, 0, stream>>>(hcat, Wout, madd, out + 1);
    loss_kernel<<<(B_ * S_ * C_ * 32) / 256, 256, 0, stream>>>(out + 1, target, accum);
    finalize_kernel<<<1, 1, 0, stream>>>(accum, out);
}